// MTL_41429254537783
// MI455X (gfx1250) — compile-verified
//
#include <hip/hip_runtime.h>
#include <hip/hip_bf16.h>
#include <math.h>

// ---------------------------------------------------------------------------
// BTSP hippocampal forward pass for MI455X (gfx1250, wave32).
// Memory-bound (~900MB traffic, AI << 1): fp32 throughout; dense mat-vecs run
// through V_WMMA_F32_16X16X4_F32 (16 rows per wave, fp32 matrix pipe); the
// near-empty CA3 vector (K=2 sparsemoid) uses a sparse gather mat-vec to skip
// a 256MB dense read. All reductions are fixed-order -> deterministic.
// ---------------------------------------------------------------------------

typedef float v2f __attribute__((ext_vector_type(2)));
typedef float v4f __attribute__((ext_vector_type(4)));
typedef float v8f __attribute__((ext_vector_type(8)));

#define BETA_F 10.0f
#define ALPHA_F 0.01f
#define NSLICE 4

// ---------------- WMMA fp32 mat-vec: part[slice][m] = W[m, kslice] @ x ------
// A tile = 16x4 fp32 weights: lane l (0..15)/(16..31) holds row m0+l%16,
//   VGPR r + half h -> K = r + 2h  => contiguous float2 load per lane.
// B tile = x broadcast over all 16 columns with the same K mapping, so every
//   column of D equals the 16-row partial mat-vec result.
__global__ void __launch_bounds__(128)
btsp_matvec_wmma(const float* __restrict__ W, const float* __restrict__ x,
                 float* __restrict__ part, int M, int Kdim)
{
    const int lane = threadIdx.x & 31;
    const int wv   = threadIdx.x >> 5;                // wave in block (0..3)
    const int m0   = (blockIdx.x * 4 + wv) * 16;      // 16 rows per wave
    const int klen = Kdim / NSLICE;
    const int k0   = blockIdx.y * klen;
    const int half = lane >> 4;
    const int l    = lane & 15;

    const float* __restrict__ wr = W + (size_t)(m0 + l) * (size_t)Kdim;

    v8f c = {0.f, 0.f, 0.f, 0.f, 0.f, 0.f, 0.f, 0.f};
#pragma unroll 8
    for (int k = k0; k < k0 + klen; k += 4) {
        v2f a = *(const v2f*)(wr + k + 2 * half);   // W[row][k+2h .. k+2h+1]
        v2f b = *(const v2f*)(x  + k + 2 * half);   // x[k+2h .. k+2h+1] (bcast cols)
        // 8 args: (neg_a, A, neg_b, B, c_mod, C, reuse_a, reuse_b)
        c = __builtin_amdgcn_wmma_f32_16x16x4_f32(
                false, a, false, b, (short)0, c, false, false);
    }

    // D layout: VGPR r, lane 0 -> (M=r, N=0); lane 16 -> (M=8+r, N=0)
    float* yp = part + (size_t)blockIdx.y * (size_t)M + m0;
    if (lane == 0) {
#pragma unroll
        for (int r = 0; r < 8; ++r) yp[r] = c[r];
    } else if (lane == 16) {
#pragma unroll
        for (int r = 0; r < 8; ++r) yp[8 + r] = c[r];
    }
}

// ---------------- fixed-order K-slice reduction -----------------------------
__global__ void btsp_reduce_part(const float* __restrict__ part,
                                 float* __restrict__ y, int M)
{
    int i = blockIdx.x * blockDim.x + threadIdx.x;
    if (i < M) {
        float a = 0.f;
#pragma unroll
        for (int s = 0; s < NSLICE; ++s) a += part[(size_t)s * M + i];
        y[i] = a;
    }
}

// ---------------- exact K-th largest via 32-pass radix select ---------------
// Monotone map fp32 -> u32, vector staged in LDS; integer counts only.
__global__ void __launch_bounds__(1024)
btsp_select_kth(const float* __restrict__ v, int n, int K, float* __restrict__ out)
{
    __shared__ unsigned su[8192];
    __shared__ unsigned cnt;
    __shared__ unsigned prefix;
    const int tid = threadIdx.x;

    for (int i = tid; i < n; i += 1024) {
        unsigned b = __float_as_uint(v[i]);
        su[i] = (b & 0x80000000u) ? ~b : (b | 0x80000000u);
    }
    if (tid == 0) prefix = 0u;
    __syncthreads();

    for (int bit = 31; bit >= 0; --bit) {
        if (tid == 0) cnt = 0u;
        __syncthreads();
        const unsigned want = prefix | (1u << bit);
        unsigned local = 0u;
        for (int i = tid; i < n; i += 1024) local += (su[i] >= want) ? 1u : 0u;
        if (local) atomicAdd(&cnt, local);
        __syncthreads();
        if (tid == 0 && cnt >= (unsigned)K) prefix = want;
        __syncthreads();
    }

    if (tid == 0) {
        const unsigned p = prefix;
        const unsigned b = (p & 0x80000000u) ? (p & 0x7FFFFFFFu) : ~p;
        *out = __uint_as_float(b);
    }
}

// ---------------- sparsemoid: sigmoid(beta*(x - t)) [+ hard mask] -----------
__global__ void btsp_sparsemoid(const float* __restrict__ v,
                                const float* __restrict__ th,
                                float* __restrict__ y, int n, int mask)
{
    int i = blockIdx.x * blockDim.x + threadIdx.x;
    if (i >= n) return;
    const float t  = *th;
    const float xv = v[i];
    float s = 1.0f / (1.0f + expf(-BETA_F * (xv - t)));
    if (mask && xv < t) s = 0.0f;
    y[i] = s;
}

// ---------------- deterministic nonzero compaction (x_ca3: ~2 nnz) ----------
__global__ void __launch_bounds__(256)
btsp_compact_nz(const float* __restrict__ x, int n,
                float* __restrict__ vals, int* __restrict__ idx,
                int* __restrict__ nnz)
{
    __shared__ float sx[8192];
    for (int i = threadIdx.x; i < n; i += blockDim.x) sx[i] = x[i];
    __syncthreads();
    if (threadIdx.x == 0) {
        int c = 0;
        for (int i = 0; i < n; ++i) {
            const float v = sx[i];
            if (v != 0.0f) { vals[c] = v; idx[c] = i; ++c; }
        }
        *nnz = c;
    }
}

// ---------------- sparse mat-vec: y = sum_t vals[t] * W[:, idx[t]] ----------
__global__ void btsp_matvec_sparse(const float* __restrict__ W,
                                   const float* __restrict__ vals,
                                   const int* __restrict__ idx,
                                   const int* __restrict__ nnzp,
                                   float* __restrict__ y, int Kdim)
{
    const int row = blockIdx.x * blockDim.x + threadIdx.x;
    const int nnz = *nnzp;
    const float* __restrict__ wr = W + (size_t)row * (size_t)Kdim;
    float acc = 0.f;
    for (int t = 0; t < nnz; ++t) acc = fmaf(vals[t], wr[idx[t]], acc);
    y[row] = acc;
}

// ---------------- Hebbian/BTSP update: Wnew = (1-a*IS)W + a*IS*x_ca3^T ------
__global__ void __launch_bounds__(256)
btsp_update_w(const float* __restrict__ W, const float* __restrict__ IS,
              const float* __restrict__ xca3, float* __restrict__ out)
{
    const size_t q = (size_t)blockIdx.x * blockDim.x + threadIdx.x; // float4 idx
    const size_t e = q * 4;
    const int row = (int)(e >> 13);       // / 8192
    const int col = (int)(e & 8191u);
    const float s = IS[row] * ALPHA_F;
    const v4f w  = *(const v4f*)(W + e);
    const v4f xc = *(const v4f*)(xca3 + col);
    v4f r;
    r.x = fmaf(s, xc.x - w.x, w.x);       // (1-s)*w + s*x
    r.y = fmaf(s, xc.y - w.y, w.y);
    r.z = fmaf(s, xc.z - w.z, w.z);
    r.w = fmaf(s, xc.w - w.w, w.w);
    *(v4f*)(out + e) = r;
}

// ---------------------------------------------------------------------------
extern "C" void kernel_launch(void* const* d_in, const int* in_sizes, int n_in,
                              void* d_out, int out_size, void* d_ws, size_t ws_size,
                              hipStream_t stream)
{
    (void)in_sizes; (void)n_in; (void)out_size; (void)ws_size;
    const float* x_ei      = (const float*)d_in[0];   // [4096]
    const float* W_ei_ca3  = (const float*)d_in[1];   // [8192,4096]
    const float* W_ei_ca1  = (const float*)d_in[2];   // [8192,4096]
    const float* W_ca3_ca1 = (const float*)d_in[3];   // [8192,8192]
    const float* W_ca1_eo  = (const float*)d_in[4];   // [4096,8192]

    float* out  = (float*)d_out;
    float* x_eo = out;            // [4096]
    float* Wnew = out + 4096;     // [8192*8192]

    // workspace carve (floats)
    float* ws     = (float*)d_ws;
    float* part   = ws;                   // NSLICE*8192 = 32768
    float* v1     = ws + 32768;           // 8192
    float* v2     = ws + 40960;           // 8192
    float* v3     = ws + 49152;           // 8192
    float* v4     = ws + 57344;           // 4096
    float* xca3   = ws + 61440;           // 8192
    float* xca1   = ws + 69632;           // 8192
    float* ISv    = ws + 77824;           // 8192
    float* th     = ws + 86016;           // 4
    float* spvals = ws + 86020;           // 8192
    int*   spidx  = (int*)(ws + 94212);   // 8192 ints
    int*   nnzp   = (int*)(ws + 102404);  // 1 int

    // 1) v1 = W_ei_ca3 @ x_ei ; x_ca3 = sparsemoid(v1, K=2, mask)
    btsp_matvec_wmma<<<dim3(8192 / 64, NSLICE), 128, 0, stream>>>(W_ei_ca3, x_ei, part, 8192, 4096);
    btsp_reduce_part<<<8192 / 256, 256, 0, stream>>>(part, v1, 8192);
    btsp_select_kth<<<1, 1024, 0, stream>>>(v1, 8192, 2, th + 0);
    btsp_sparsemoid<<<8192 / 256, 256, 0, stream>>>(v1, th + 0, xca3, 8192, 1);

    // 2) v2 = W_ca3_ca1 @ x_ca3 (sparse: ~2 nnz) ; x_ca1 = sparsemoid(v2, K=100, no mask)
    btsp_compact_nz<<<1, 256, 0, stream>>>(xca3, 8192, spvals, spidx, nnzp);
    btsp_matvec_sparse<<<8192 / 256, 256, 0, stream>>>(W_ca3_ca1, spvals, spidx, nnzp, v2, 8192);
    btsp_select_kth<<<1, 1024, 0, stream>>>(v2, 8192, 100, th + 1);
    btsp_sparsemoid<<<8192 / 256, 256, 0, stream>>>(v2, th + 1, xca1, 8192, 0);

    // 3) v3 = W_ei_ca1 @ x_ei ; IS = sparsemoid(v3, K=100, mask)
    btsp_matvec_wmma<<<dim3(8192 / 64, NSLICE), 128, 0, stream>>>(W_ei_ca1, x_ei, part, 8192, 4096);
    btsp_reduce_part<<<8192 / 256, 256, 0, stream>>>(part, v3, 8192);
    btsp_select_kth<<<1, 1024, 0, stream>>>(v3, 8192, 100, th + 2);
    btsp_sparsemoid<<<8192 / 256, 256, 0, stream>>>(v3, th + 2, ISv, 8192, 1);

    // 4) v4 = W_ca1_eo @ x_ca1 ; x_eo = sparsemoid(v4, K=50, mask)
    btsp_matvec_wmma<<<dim3(4096 / 64, NSLICE), 128, 0, stream>>>(W_ca1_eo, xca1, part, 4096, 8192);
    btsp_reduce_part<<<4096 / 256, 256, 0, stream>>>(part, v4, 4096);
    btsp_select_kth<<<1, 1024, 0, stream>>>(v4, 4096, 50, th + 3);
    btsp_sparsemoid<<<4096 / 256, 256, 0, stream>>>(v4, th + 3, x_eo, 4096, 1);

    // 5) Wnew = (1 - IS*alpha)*W_ca3_ca1 + alpha*(IS @ x_ca3^T)  [512MB stream]
    btsp_update_w<<<(8192u * 8192u / 4u) / 256u, 256, 0, stream>>>(W_ca3_ca1, ISv, xca3, Wnew);
}